// CElegansModel_Torch_55757265436779
// MI455X (gfx1250) — compile-verified
//
#include <hip/hip_runtime.h>

typedef __attribute__((ext_vector_type(2))) float v2f;
typedef __attribute__((ext_vector_type(8))) float v8f;

#define T_LEN 2048
#define BATCH 4096
#define HID   8

__device__ __forceinline__ float fast_rcp(float x) {
    return __builtin_amdgcn_rcpf(x);               // raw v_rcp_f32
}

__device__ __forceinline__ float fast_tanh(float x) {
#if __has_builtin(__builtin_amdgcn_tanhf)
    return __builtin_amdgcn_tanhf(x);              // hw v_tanh_f32 (CDNA5 TRANS op)
#else
    float e = __expf(2.0f * x);
    return 1.0f - 2.0f * fast_rcp(e + 1.0f);
#endif
}

__device__ __forceinline__ float fast_sigmoid(float x) {
#if __has_builtin(__builtin_amdgcn_tanhf)
    return fmaf(0.5f, __builtin_amdgcn_tanhf(0.5f * x), 0.5f);  // 3 ops, no exp/div
#else
    return fast_rcp(1.0f + __expf(-x));
#endif
}

// lane <-> lane^8 exchange: DPP row_ror:8 on a 16-lane row == XOR 8 (1 VALU op,
// no LDS pipe), instead of ds_bpermute.
__device__ __forceinline__ float lane_xor8(float v) {
#if __has_builtin(__builtin_amdgcn_update_dpp)
    int i = __float_as_int(v);
    i = __builtin_amdgcn_update_dpp(0, i, 0x128 /*row_ror:8*/, 0xf, 0xf, true);
    return __int_as_float(i);
#elif __has_builtin(__builtin_amdgcn_mov_dpp)
    int i = __float_as_int(v);
    i = __builtin_amdgcn_mov_dpp(i, 0x128 /*row_ror:8*/, 0xf, 0xf, true);
    return __int_as_float(i);
#else
    return __shfl_xor(v, 8, 32);
#endif
}

// Same-wave scheduling fence: LDS ops from one wave are executed in order by
// the LDS pipe, so the D->A transpose store->load only needs compiler ordering.
__device__ __forceinline__ void wave_lds_fence() {
    __builtin_amdgcn_wave_barrier();               // emits no instruction
}

// One wave (32 threads) per 16-batch tile. 256 blocks x 32 threads.
__global__ __launch_bounds__(32)
void gru_wmma_kernel(const float* __restrict__ x,
                     const float* __restrict__ W_ih,
                     const float* __restrict__ W_hh,
                     const float* __restrict__ b_ih,
                     const float* __restrict__ b_hh,
                     const float* __restrict__ W_fc,
                     const float* __restrict__ b_fc,
                     float* __restrict__ out)
{
    __shared__ float sh[16 * HID];           // 16x8 h tile for D->A transpose

    const int lane = threadIdx.x;            // 0..31 (wave32)
    const int n16  = lane & 15;              // N column (B/C/D) or M row (A)
    const int half = lane >> 4;              // 0/1 -> K chunk select
    const int ka   = half * 2;               // A/B layout: K = ka, ka+1 per VGPR pair
    const int b0   = blockIdx.x * 16;

    // ---------------- constant B operands (weights), per-lane -------------
    v2f z2; z2.x = 0.f; z2.y = 0.f;
    v2f Bih_rz = z2, Bih_n = z2, Bhh_rz_a = z2, Bhh_rz_b = z2;
    v2f Bhnfc_a = z2, Bhnfc_b = z2;

    // tile0 columns 0..15: gates r (0..7) and z (8..15)
    Bih_rz.x   = W_ih[n16*4 + ka];       Bih_rz.y   = W_ih[n16*4 + ka + 1];
    Bhh_rz_a.x = W_hh[n16*8 + ka];       Bhh_rz_a.y = W_hh[n16*8 + ka + 1];
    Bhh_rz_b.x = W_hh[n16*8 + 4 + ka];   Bhh_rz_b.y = W_hh[n16*8 + 4 + ka + 1];
    // tile1: columns 0..7 = n gate, columns 8..11 = W_fc (fused FC output)
    if (n16 < 8) {
        Bih_n.x   = W_ih[(16+n16)*4 + ka];     Bih_n.y   = W_ih[(16+n16)*4 + ka + 1];
        Bhnfc_a.x = W_hh[(16+n16)*8 + ka];     Bhnfc_a.y = W_hh[(16+n16)*8 + ka + 1];
        Bhnfc_b.x = W_hh[(16+n16)*8 + 4 + ka]; Bhnfc_b.y = W_hh[(16+n16)*8 + 4 + ka + 1];
    } else if (n16 < 12) {
        Bhnfc_a.x = W_fc[(n16-8)*8 + ka];      Bhnfc_a.y = W_fc[(n16-8)*8 + ka + 1];
        Bhnfc_b.x = W_fc[(n16-8)*8 + 4 + ka];  Bhnfc_b.y = W_fc[(n16-8)*8 + 4 + ka + 1];
    }

    // ---------------- bias broadcasts as C operands ------------------------
    float crz = b_ih[n16] + b_hh[n16];                       // r,z: both biases fold
    float cxn = (n16 < 8) ? b_ih[16 + n16] : 0.0f;           // xn bias
    float chn = (n16 < 8) ? b_hh[16 + n16]                   // hn bias
              : ((n16 < 12) ? b_fc[n16 - 8] : 0.0f);         // fused FC bias
    v8f Crz, Cxn, Chn;
    #pragma unroll
    for (int i = 0; i < 8; ++i) { Crz[i] = crz; Cxn[i] = cxn; Chn[i] = chn; }

    // ---------------- streaming pointers -----------------------------------
    // x[b][t][0..3]; A-layout load: lane n16 = row m, components K=ka,ka+1
    const v2f* xp = (const v2f*)(x + (size_t)(b0 + n16) * T_LEN * 4);
    const bool o_lane = (n16 >= 8) && (n16 < 12);
    float* outp = out + (size_t)(b0 + 8*half) * T_LEN * 4 + (o_lane ? (n16 - 8) : 0);

    // ---------------- state -------------------------------------------------
    v8f hD;                                  // h in C/D layout (valid lanes n16<8)
    #pragma unroll
    for (int i = 0; i < 8; ++i) hD[i] = 0.0f;
    v2f ha0 = z2, ha1 = z2;                  // h in A layout, K chunks 0..3 / 4..7
    v2f xa  = xp[half];                      // x_0 in A layout

    for (int t = 0; t < T_LEN; ++t) {
        // prefetch the x stream
        if (t + 64 < T_LEN) __builtin_prefetch((const void*)(xp + 2*(t+64) + half), 0, 1);
        v2f xan = z2;
        if (t + 1 < T_LEN) xan = xp[2*(t+1) + half];

        // tile0: (xr|xz) + (hr|hz) + biases      D = A*B + C chained
        v8f g0 = __builtin_amdgcn_wmma_f32_16x16x4_f32(false, xa,  false, Bih_rz,   (short)0, Crz, false, false);
        g0     = __builtin_amdgcn_wmma_f32_16x16x4_f32(false, ha0, false, Bhh_rz_a, (short)0, g0,  false, false);
        g0     = __builtin_amdgcn_wmma_f32_16x16x4_f32(false, ha1, false, Bhh_rz_b, (short)0, g0,  false, false);
        // tile1a: xn (kept separate: n = tanh(xn + r*hn))
        v8f t1x = __builtin_amdgcn_wmma_f32_16x16x4_f32(false, xa,  false, Bih_n,   (short)0, Cxn, false, false);
        // tile1b: hn (cols 0..7) and fused out[t-1] = W_fc*h_{t-1}+b_fc (cols 8..11)
        v8f t1h = __builtin_amdgcn_wmma_f32_16x16x4_f32(false, ha0, false, Bhnfc_a, (short)0, Chn, false, false);
        t1h     = __builtin_amdgcn_wmma_f32_16x16x4_f32(false, ha1, false, Bhnfc_b, (short)0, t1h, false, false);

        // emit out[b, t-1, o] from fused columns (stores write-combine in L2)
        if (t > 0 && o_lane) {
            #pragma unroll
            for (int r = 0; r < 8; ++r)
                outp[((size_t)r * T_LEN + (t - 1)) * 4] = t1h[r];
        }

        // gate math (valid where n16<8; one sigmoid covers both r and z)
        v8f hN;
        #pragma unroll
        for (int r = 0; r < 8; ++r) {
            float s  = fast_sigmoid(g0[r]);        // r (n16<8) / z (n16>=8)
            float zz = lane_xor8(s);               // z aligned to r lanes (1 DPP op)
            float np = t1x[r] + s * t1h[r];        // xn + r*hn
            float nv = fast_tanh(np);
            hN[r] = nv + zz * (hD[r] - nv);        // (1-z)*n + z*h
        }
        hD = hN;

        // D-layout -> LDS -> A-layout for next step.
        // Same-wave LDS ops execute in order in the LDS pipe; only compiler
        // ordering is needed (wave_lds_fence emits no instruction).
        if (n16 < 8) {
            #pragma unroll
            for (int r = 0; r < 8; ++r)
                sh[(r + 8*half) * HID + n16] = hN[r];
        }
        wave_lds_fence();
        ha0 = *(const v2f*)&sh[n16 * HID + ka];
        ha1 = *(const v2f*)&sh[n16 * HID + 4 + ka];
        wave_lds_fence();

        xa = xan;
    }

    // epilogue: out[T-1] = W_fc * h_{T-1} + b_fc via the same fused tile
    v8f of = __builtin_amdgcn_wmma_f32_16x16x4_f32(false, ha0, false, Bhnfc_a, (short)0, Chn, false, false);
    of     = __builtin_amdgcn_wmma_f32_16x16x4_f32(false, ha1, false, Bhnfc_b, (short)0, of,  false, false);
    if (o_lane) {
        #pragma unroll
        for (int r = 0; r < 8; ++r)
            outp[((size_t)r * T_LEN + (T_LEN - 1)) * 4] = of[r];
    }
}

extern "C" void kernel_launch(void* const* d_in, const int* in_sizes, int n_in,
                              void* d_out, int out_size, void* d_ws, size_t ws_size,
                              hipStream_t stream) {
    (void)in_sizes; (void)n_in; (void)out_size; (void)d_ws; (void)ws_size;
    const float* x    = (const float*)d_in[0];
    const float* W_ih = (const float*)d_in[1];
    const float* W_hh = (const float*)d_in[2];
    const float* b_ih = (const float*)d_in[3];
    const float* b_hh = (const float*)d_in[4];
    const float* W_fc = (const float*)d_in[5];
    const float* b_fc = (const float*)d_in[6];
    float* out = (float*)d_out;

    dim3 grid(BATCH / 16), block(32);
    gru_wmma_kernel<<<grid, block, 0, stream>>>(x, W_ih, W_hh, b_ih, b_hh, W_fc, b_fc, out);
}